// KPConvLayer_40991167873570
// MI455X (gfx1250) — compile-verified
//
#include <hip/hip_runtime.h>

typedef __attribute__((ext_vector_type(2))) float v2f;
typedef __attribute__((ext_vector_type(8))) float v8f;

#define NPTS_  16384
#define KNBR_  16
#define PKER_  19
#define DIN_   64
#define DOUT_  320

// One block = 16 points (one WMMA M-tile), 128 threads = 4 waves.
// Wave w computes n-tiles [w*5, w*5+5) of the 20 tiles covering Dout=320.
__global__ __launch_bounds__(128) void kpconv_fused_kernel(
    const float* __restrict__ X,    // [B, N, 3]
    const float* __restrict__ F,    // [B, N, 64]
    const int*   __restrict__ NBR,  // [B, N, 16]
    const float* __restrict__ Q,    // [19, 3]
    const float* __restrict__ W,    // [64, 320]
    float*       __restrict__ OUT)  // [B, N, 320]
{
    __shared__ float sQ[PKER_ * 3];
    __shared__ float sCX[16 * 3];
    __shared__ int   sIdx[16 * KNBR_];
    __shared__ float sS[16 * KNBR_];
    __shared__ float sG[16 * DIN_];   // A-matrix tile: 16 points x 64 dims, f32

    const int t    = threadIdx.x;
    const int base = blockIdx.x * 16;           // global flat point index base (b*N+n)

    // --- stage 0: stage kernel points + center coordinates into LDS ---
    if (t < PKER_ * 3) sQ[t] = Q[t];
    if (t < 16) {
        const int gp = base + t;
        sCX[t * 3 + 0] = X[(size_t)gp * 3 + 0];
        sCX[t * 3 + 1] = X[(size_t)gp * 3 + 1];
        sCX[t * 3 + 2] = X[(size_t)gp * 3 + 2];
    }
    __syncthreads();

    // --- stage 1: per-(point,neighbor) scalar weight s = sum_p relu(1 - |rel - q_p|) ---
    for (int pair = t; pair < 16 * KNBR_; pair += 128) {
        const int p  = pair >> 4;
        const int k  = pair & 15;
        const int gp = base + p;
        const int b  = gp >> 14;                // / NPTS_
        const int nbr = NBR[(size_t)gp * KNBR_ + k];
        sIdx[pair] = nbr;
        const float* nx = X + (size_t)(b * NPTS_ + nbr) * 3;
        const float rx = nx[0] - sCX[p * 3 + 0];
        const float ry = nx[1] - sCX[p * 3 + 1];
        const float rz = nx[2] - sCX[p * 3 + 2];
        float s = 0.0f;
        #pragma unroll
        for (int q = 0; q < PKER_; ++q) {
            const float dx = rx - sQ[q * 3 + 0];
            const float dy = ry - sQ[q * 3 + 1];
            const float dz = rz - sQ[q * 3 + 2];
            const float d  = sqrtf(dx * dx + dy * dy + dz * dz);
            s += fmaxf(0.0f, 1.0f - d);         // SIGMA = 1.0
        }
        sS[pair] = s;
    }
    __syncthreads();

    // --- stage 2: G[p][d] = sum_k s[p][k] * F[nbr(p,k)][d]  -> LDS ---
    {
        const int p  = t >> 3;                  // 16 points
        const int dg = (t & 7) * 8;             // 8 threads/point x 8 dims
        const int gp = base + p;
        const int b  = gp >> 14;
        float a0 = 0, a1 = 0, a2 = 0, a3 = 0, a4 = 0, a5 = 0, a6 = 0, a7 = 0;
        #pragma unroll
        for (int k = 0; k < KNBR_; ++k) {
            const int   nbr = sIdx[p * KNBR_ + k];
            const float s   = sS[p * KNBR_ + k];
            const float4* fp = reinterpret_cast<const float4*>(
                F + (size_t)(b * NPTS_ + nbr) * DIN_ + dg);
            const float4 f0 = fp[0];
            const float4 f1 = fp[1];
            a0 += s * f0.x; a1 += s * f0.y; a2 += s * f0.z; a3 += s * f0.w;
            a4 += s * f1.x; a5 += s * f1.y; a6 += s * f1.z; a7 += s * f1.w;
        }
        float* g = sG + p * DIN_ + dg;
        g[0] = a0; g[1] = a1; g[2] = a2; g[3] = a3;
        g[4] = a4; g[5] = a5; g[6] = a6; g[7] = a7;
    }
    __syncthreads();

    // --- stage 3: OUT[16 x 320] = G[16 x 64] @ W[64 x 320] via v_wmma_f32_16x16x4_f32 ---
    {
        const int lane = t & 31;
        const int wv   = t >> 5;                 // wave id 0..3
        const int m    = lane & 15;
        const int koff = (lane < 16) ? 0 : 2;    // A/B K placement per ISA 7.12.2
        const int c0   = (wv * 5) * 16 + m;      // first column this lane touches

        v8f acc0 = {}, acc1 = {}, acc2 = {}, acc3 = {}, acc4 = {};

        #pragma unroll
        for (int ks = 0; ks < 16; ++ks) {        // K=64 in steps of 4
            const int r = ks * 4 + koff;
            // A fragment: lane m holds G[m][r], G[m][r+1]  (ds_load_b64)
            const float* gp = sG + m * DIN_ + r;
            v2f a;
            a.x = gp[0];
            a.y = gp[1];
            // B fragments: W rows r, r+1 at 5 columns (coalesced dword loads, L2-hot)
            const float* w0 = W + (size_t)r * DOUT_;
            const float* w1 = W + (size_t)(r + 1) * DOUT_;
            v2f b0; b0.x = w0[c0 +  0]; b0.y = w1[c0 +  0];
            v2f b1; b1.x = w0[c0 + 16]; b1.y = w1[c0 + 16];
            v2f b2; b2.x = w0[c0 + 32]; b2.y = w1[c0 + 32];
            v2f b3; b3.x = w0[c0 + 48]; b3.y = w1[c0 + 48];
            v2f b4; b4.x = w0[c0 + 64]; b4.y = w1[c0 + 64];

            acc0 = __builtin_amdgcn_wmma_f32_16x16x4_f32(false, a, false, b0, (short)0, acc0, false, false);
            acc1 = __builtin_amdgcn_wmma_f32_16x16x4_f32(false, a, false, b1, (short)0, acc1, false, false);
            acc2 = __builtin_amdgcn_wmma_f32_16x16x4_f32(false, a, false, b2, (short)0, acc2, false, false);
            acc3 = __builtin_amdgcn_wmma_f32_16x16x4_f32(false, a, false, b3, (short)0, acc3, false, false);
            acc4 = __builtin_amdgcn_wmma_f32_16x16x4_f32(false, a, false, b4, (short)0, acc4, false, false);
        }

        // D layout: VGPR v -> row M = v + (lane<16 ? 0 : 8), col N = lane&15
        const int mrow = (lane < 16) ? 0 : 8;
        #pragma unroll
        for (int v = 0; v < 8; ++v) {
            const size_t row = (size_t)(base + mrow + v) * DOUT_;
            OUT[row + c0 +  0] = acc0[v];
            OUT[row + c0 + 16] = acc1[v];
            OUT[row + c0 + 32] = acc2[v];
            OUT[row + c0 + 48] = acc3[v];
            OUT[row + c0 + 64] = acc4[v];
        }
    }
}

extern "C" void kernel_launch(void* const* d_in, const int* in_sizes, int n_in,
                              void* d_out, int out_size, void* d_ws, size_t ws_size,
                              hipStream_t stream) {
    (void)in_sizes; (void)n_in; (void)d_ws; (void)ws_size; (void)out_size;
    const float* X   = (const float*)d_in[0];   // [4,16384,3]
    const float* F   = (const float*)d_in[1];   // [4,16384,64]
    const int*   NBR = (const int*)  d_in[2];   // [4,16384,16]
    const float* Q   = (const float*)d_in[3];   // [19,3]
    const float* W   = (const float*)d_in[4];   // [64,320]
    float*       OUT = (float*)d_out;           // [4,16384,320]

    const int total_pts = 4 * NPTS_;            // 65536
    dim3 grid(total_pts / 16);                  // 4096 blocks, 16 points each
    dim3 block(128);                            // 4 waves (wave32)
    kpconv_fused_kernel<<<grid, block, 0, stream>>>(X, F, NBR, Q, W, OUT);
}